// GeneralisedMessagePassing_2267742732505
// MI455X (gfx1250) — compile-verified
//
#include <hip/hip_runtime.h>

typedef __attribute__((ext_vector_type(2))) float v2f;
typedef __attribute__((ext_vector_type(8))) float v8f;

#define NUM_V 100000
#define NUM_R 1000
#define DIMD  128
#define DIMI  64
#define DIMH  128
#define NB    4096
#define NN    64

// ---------------------------------------------------------------------------
// Phase 1: Y[nrows x 128] = X[nrows x K] @ W[K x 128] + bias[128]
// One wave computes a 16x128 tile via V_WMMA_F32_16X16X4_F32 (exact fp32).
// A-fragment (16x4 f32): lane m=l&15 holds row m; hh=l>>4 selects K-pair
//   {k, k+1} (hh=0) or {k+2, k+3} (hh=1)  -> contiguous float2 load from X.
// B-fragment (4x16 f32): lane n=l&15 holds col n; same K-pair split.
// C/D (16x16 f32): VGPR j, lane l -> (row0 + j + 8*(l>>4), 16*t + (l&15)).
// ---------------------------------------------------------------------------
template <int K>
__global__ __launch_bounds__(128) void gmp_proj_rows_wmma(
    const float* __restrict__ X, const float* __restrict__ W,
    const float* __restrict__ bias, float* __restrict__ Y, int nrows)
{
  const int wave = threadIdx.x >> 5;
  const int lane = threadIdx.x & 31;
  const int m    = lane & 15;
  const int hh   = lane >> 4;                 // K-pair select within fragment
  const int row0 = (blockIdx.x * 4 + wave) * 16;
  if (row0 >= nrows) return;                  // wave-uniform: EXEC stays full

  int arow = row0 + m;
  if (arow >= nrows) arow = nrows - 1;        // clamp A reads; dropped on store

  v8f acc[8];
#pragma unroll
  for (int t = 0; t < 8; ++t)
    acc[t] = (v8f){0.f, 0.f, 0.f, 0.f, 0.f, 0.f, 0.f, 0.f};

  const float* xrow = X + (size_t)arow * K;
  for (int k = 0; k < K; k += 4) {
    v2f a;
    a.x = xrow[k + 2 * hh];
    a.y = xrow[k + 2 * hh + 1];
    const float* w0 = W + (size_t)(k + 2 * hh) * DIMH;
    const float* w1 = W + (size_t)(k + 2 * hh + 1) * DIMH;
#pragma unroll
    for (int t = 0; t < 8; ++t) {
      const int n = t * 16 + m;
      v2f b;
      b.x = w0[n];
      b.y = w1[n];
      acc[t] = __builtin_amdgcn_wmma_f32_16x16x4_f32(
          /*neg_a=*/false, a, /*neg_b=*/false, b,
          /*c_mod=*/(short)0, acc[t], /*reuse_a=*/false, /*reuse_b=*/false);
    }
  }

#pragma unroll
  for (int t = 0; t < 8; ++t) {
    const int n = t * 16 + m;
    const float bv = bias[n];
#pragma unroll
    for (int j = 0; j < 8; ++j) {
      const int rr = row0 + j + hh * 8;
      if (rr < nrows) Y[(size_t)rr * DIMH + n] = acc[t][j] + bv;
    }
  }
}

// ---------------------------------------------------------------------------
// Phase 1c: pp[h] = prod_a ( (P[a] @ WP)[h] + bP[h] )   (tiny: 1 block)
// ---------------------------------------------------------------------------
__global__ __launch_bounds__(128) void gmp_compute_pp(
    const float* __restrict__ P, const float* __restrict__ WP,
    const float* __restrict__ bP, float* __restrict__ pp)
{
  const int h = threadIdx.x;
  float prod = 1.0f;
#pragma unroll
  for (int a = 0; a < 3; ++a) {
    float s = bP[h];
    for (int k = 0; k < DIMD; ++k)
      s = fmaf(P[a * DIMD + k], WP[(size_t)k * DIMH + h], s);
    prod *= s;
  }
  pp[h] = prod;
}

// ---------------------------------------------------------------------------
// Phase 2: out[b,h] = pp[h] * sum_n Rp[r[b,n]][h] * Vp[e0][h]*Vp[e1][h]*Vp[e2][h]
// One block per b; thread h owns one output lane. Tables (52 MB) sit in L2.
// ---------------------------------------------------------------------------
__global__ __launch_bounds__(128) void gmp_gather_reduce(
    const int* __restrict__ r, const int* __restrict__ e,
    const float* __restrict__ Rp, const float* __restrict__ Vp,
    const float* __restrict__ pp, float* __restrict__ out)
{
  const int b = blockIdx.x;
  const int h = threadIdx.x;
  const int* rb = r + (size_t)b * NN;
  const int* e0 = e + 0 * (size_t)NB * NN + (size_t)b * NN;
  const int* e1 = e + 1 * (size_t)NB * NN + (size_t)b * NN;
  const int* e2 = e + 2 * (size_t)NB * NN + (size_t)b * NN;

  float acc = 0.0f;
#pragma unroll 4
  for (int n = 0; n < NN; ++n) {
    const int ri = rb[n];
    const int i0 = e0[n];
    const int i1 = e1[n];
    const int i2 = e2[n];
    float x = Rp[(size_t)ri * DIMH + h];
    x *= Vp[(size_t)i0 * DIMH + h];
    x *= Vp[(size_t)i1 * DIMH + h];
    x *= Vp[(size_t)i2 * DIMH + h];
    acc += x;
  }
  out[(size_t)b * DIMH + h] = acc * pp[h];
}

extern "C" void kernel_launch(void* const* d_in, const int* in_sizes, int n_in,
                              void* d_out, int out_size, void* d_ws, size_t ws_size,
                              hipStream_t stream)
{
  const int*   r  = (const int*)  d_in[0];
  const int*   e  = (const int*)  d_in[1];
  const float* V  = (const float*)d_in[2];
  const float* R  = (const float*)d_in[3];
  const float* P  = (const float*)d_in[4];
  const float* WV = (const float*)d_in[5];
  const float* bV = (const float*)d_in[6];
  const float* WR = (const float*)d_in[7];
  const float* bR = (const float*)d_in[8];
  const float* WP = (const float*)d_in[9];
  const float* bP = (const float*)d_in[10];
  float* out = (float*)d_out;

  // Workspace layout (floats): Vp[100000*128] | Rp[1000*128] | pp[128]
  float* Vp = (float*)d_ws;
  float* Rp = Vp + (size_t)NUM_V * DIMH;
  float* pp = Rp + (size_t)NUM_R * DIMH;

  // Phase 1: project the tables once (GEMMs on the fp32 WMMA path)
  gmp_proj_rows_wmma<DIMD + DIMI><<<(NUM_V + 63) / 64, 128, 0, stream>>>(
      V, WV, bV, Vp, NUM_V);
  gmp_proj_rows_wmma<DIMD><<<(NUM_R + 63) / 64, 128, 0, stream>>>(
      R, WR, bR, Rp, NUM_R);
  gmp_compute_pp<<<1, 128, 0, stream>>>(P, WP, bP, pp);

  // Phase 2: gather-multiply-reduce out of L2
  gmp_gather_reduce<<<NB, 128, 0, stream>>>(r, e, Rp, Vp, pp, out);
}